// Model_17746804867087
// MI455X (gfx1250) — compile-verified
//
#include <hip/hip_runtime.h>

typedef __attribute__((ext_vector_type(2))) float v2f;
typedef __attribute__((ext_vector_type(8))) float v8f;

extern __shared__ float smem[];

// ---------------------------------------------------------------------------
// GEMM: out[M,NOUT] = (RELU ? max(A,0) : A)[M,KTOT] @ W[KTOT,NOUT] + bias
//
// Block = 256 threads = 8 waves; wave w owns rows [blk*128 + w*16, +16) and
// computes the full NOUT width as NOUT/16 WMMA f32 16x16x4 tiles over K.
//
// W is staged into LDS K-pair interleaved: pair-row p (= k/2) holds
// {W[2p][n], W[2p+1][n]} contiguously at smem[p*PROW + n*2], so each lane's
// B fragment is a single aligned ds_load_b64 into an even VGPR pair.
// PROW = 2*NOUT + 32 pads consecutive pair-rows 32 banks apart: lanes 0-15
// (pair-row p) use banks 0-31, lanes 16-31 (pair-row p+1) banks 32-63 ->
// conflict-free full-wave b64 LDS reads.
//
// A fragment (ISA 7.12.2, 32-bit A 16x4): lane L holds row M=L%16, K-pair
// selected by L>>4; loaded as one global b64, software-pipelined one K-step
// ahead so the load overlaps the previous 8-WMMA group.
// C/D layout: elem (vgpr r, lane L) -> M = r + 8*(L>>4), N = L&15.
// ---------------------------------------------------------------------------
template <int KTOT, int NOUT, bool RELU>
__global__ __launch_bounds__(256) void gemm_bias_wmma(
    const float* __restrict__ A, const float* __restrict__ W,
    const float* __restrict__ bias, float* __restrict__ out, int M) {
  constexpr int KCHUNK = (KTOT < 128) ? KTOT : 128;
  constexpr int NT = NOUT / 16;
  constexpr int PROW = 2 * NOUT + 32;  // floats per pair-row (padded)

  const int lane = threadIdx.x & 31;
  const int wave = threadIdx.x >> 5;
  const int l16 = lane & 15;
  const int kh = lane >> 4;  // which K-pair of the 4-wide K step
  const int m0 = blockIdx.x * 128 + wave * 16;

  int mrow = m0 + l16;
  if (mrow >= M) mrow = M - 1;  // clamp: EXEC must stay all-1s for WMMA
  const float* arow = A + (size_t)mrow * KTOT;

  v8f acc[NT];
#pragma unroll
  for (int nt = 0; nt < NT; ++nt) acc[nt] = (v8f)0.f;

  const float* bbase = smem + (size_t)kh * PROW + l16 * 2;

  for (int kc = 0; kc < KTOT; kc += KCHUNK) {
    // ---- stage W chunk into LDS, K-pair interleaved -----------------------
    for (int i = threadIdx.x; i < (KCHUNK / 2) * (NOUT / 2); i += 256) {
      const int p = i / (NOUT / 2);
      const int nn = (i % (NOUT / 2)) * 2;
      const int k = kc + p * 2;
      const float2 r0 = *(const float2*)(W + (size_t)k * NOUT + nn);
      const float2 r1 = *(const float2*)(W + (size_t)(k + 1) * NOUT + nn);
      float* s = smem + (size_t)p * PROW + nn * 2;
      ((float2*)s)[0] = make_float2(r0.x, r1.x);
      ((float2*)s)[1] = make_float2(r0.y, r1.y);
    }
    __syncthreads();

    // ---- K loop: 8 WMMAs per 4-wide K step, A-load pipelined --------------
    float2 av = *(const float2*)(arow + kc + 2 * kh);
#pragma unroll 8
    for (int k0 = 0; k0 < KCHUNK; k0 += 4) {
      const int knext = (k0 + 4 < KCHUNK) ? (k0 + 4) : k0;
      const float2 avn = *(const float2*)(arow + kc + knext + 2 * kh);

      v2f a;
      a.x = RELU ? fmaxf(av.x, 0.f) : av.x;
      a.y = RELU ? fmaxf(av.y, 0.f) : av.y;

      const float* bp = bbase + (size_t)(k0 / 2) * PROW;
#pragma unroll
      for (int nt = 0; nt < NT; ++nt) {
        const v2f b = *(const v2f*)(bp + nt * 32);
        acc[nt] = __builtin_amdgcn_wmma_f32_16x16x4_f32(
            false, a, false, b, (short)0, acc[nt], false, false);
      }
      av = avn;
    }
    __syncthreads();
  }

  // ---- store with bias; mask tail rows (stores may be predicated) ---------
#pragma unroll
  for (int nt = 0; nt < NT; ++nt) {
    const int n = nt * 16 + l16;
    const float bv = bias[n];
#pragma unroll
    for (int r = 0; r < 8; ++r) {
      const int m = m0 + r + 8 * kh;
      if (m < M) out[(size_t)m * NOUT + n] = acc[nt][r] + bv;
    }
  }
}

// ---------------------------------------------------------------------------
// Edge scatter: out[dst[e]] += P[src[e]] * w[e].  D/4 threads per edge; each
// thread moves a float4 and issues 4 hardware global_atomic_add_f32. The
// node-feature tensors (25.6MB) are L2-resident (192MB L2), so both the
// gather reads and the scatter atomics run at L2 rates, not HBM.
// ---------------------------------------------------------------------------
template <int D>
__global__ __launch_bounds__(256) void edge_scatter(
    const float* __restrict__ P, const int* __restrict__ src,
    const int* __restrict__ dst, const float* __restrict__ w,
    float* __restrict__ out, int E) {
  constexpr int TPE = D / 4;
  const long long idx = (long long)blockIdx.x * blockDim.x + threadIdx.x;
  const int e = (int)(idx / TPE);
  if (e >= E) return;
  const int f = (int)(idx % TPE) * 4;
  const int s = src[e];
  const int d = dst[e];
  const float wt = w[e];
  const float4 v = *(const float4*)(P + (size_t)s * D + f);
  float* o = out + (size_t)d * D + f;
  unsafeAtomicAdd(o + 0, v.x * wt);
  unsafeAtomicAdd(o + 1, v.y * wt);
  unsafeAtomicAdd(o + 2, v.z * wt);
  unsafeAtomicAdd(o + 3, v.w * wt);
}

__global__ __launch_bounds__(256) void zero_f4(float* __restrict__ p,
                                               long long n4) {
  const long long i = (long long)blockIdx.x * blockDim.x + threadIdx.x;
  if (i < n4) ((float4*)p)[i] = make_float4(0.f, 0.f, 0.f, 0.f);
}

extern "C" void kernel_launch(void* const* d_in, const int* in_sizes, int n_in,
                              void* d_out, int out_size, void* d_ws,
                              size_t ws_size, hipStream_t stream) {
  const float* x = (const float*)d_in[0];
  const int* src = (const int*)d_in[1];
  const int* dst = (const int*)d_in[2];
  const float* w = (const float*)d_in[3];
  const float* W1 = (const float*)d_in[4];
  const float* b1 = (const float*)d_in[5];
  const float* W2 = (const float*)d_in[6];
  const float* b2 = (const float*)d_in[7];
  const float* W3 = (const float*)d_in[8];
  const float* b3 = (const float*)d_in[9];
  float* out = (float*)d_out;

  const int N = in_sizes[0] / 256;  // 50000 nodes
  const int E = in_sizes[1];        // 800000 edges

  float* P = (float*)d_ws;           // projected features [N,128]
  float* Agg = P + (size_t)N * 128;  // aggregated features [N,128]

  const dim3 blk(256);
  const int gemmGrid = (N + 127) / 128;
  constexpr size_t lds128 = (size_t)64 * (2 * 128 + 32) * 4;  // 73728 B
  constexpr size_t lds64 = (size_t)64 * (2 * 64 + 32) * 4;    // 40960 B

  auto zero = [&](float* p, long long n) {
    const long long n4 = n / 4;
    zero_f4<<<(unsigned)((n4 + 255) / 256), blk, 0, stream>>>(p, n4);
  };

  // ---- layer 1: P = x @ W1 + b1 ; Agg = segsum(P[src]*w -> dst)
  gemm_bias_wmma<256, 128, false>
      <<<gemmGrid, blk, lds128, stream>>>(x, W1, b1, P, N);
  zero(Agg, (long long)N * 128);
  {
    const long long t = (long long)E * 32;
    edge_scatter<128>
        <<<(unsigned)((t + 255) / 256), blk, 0, stream>>>(P, src, dst, w, Agg, E);
  }

  // ---- layer 2: P = relu(Agg) @ W2 + b2 ; Agg = segsum(...)
  gemm_bias_wmma<128, 128, true>
      <<<gemmGrid, blk, lds128, stream>>>(Agg, W2, b2, P, N);
  zero(Agg, (long long)N * 128);
  {
    const long long t = (long long)E * 32;
    edge_scatter<128>
        <<<(unsigned)((t + 255) / 256), blk, 0, stream>>>(P, src, dst, w, Agg, E);
  }

  // ---- layer 3: P = relu(Agg) @ W3 + b3 ; out = segsum(...)
  gemm_bias_wmma<128, 64, true>
      <<<gemmGrid, blk, lds64, stream>>>(Agg, W3, b3, P, N);
  zero(out, (long long)N * 64);
  {
    const long long t = (long long)E * 16;
    edge_scatter<64>
        <<<(unsigned)((t + 255) / 256), blk, 0, stream>>>(P, src, dst, w, out, E);
  }
}